// LIFLayer_17059610100048
// MI455X (gfx1250) — compile-verified
//
#include <hip/hip_runtime.h>
#include <cmath>

#ifndef __has_builtin
#define __has_builtin(x) 0
#endif

// ---- CDNA5 async LDS->global path (guarded; fallback = synchronous stores) ----
#if defined(__AMDGCN__) && __has_builtin(__builtin_amdgcn_global_store_async_from_lds_b128)
#define HAVE_ASYNC_LDS 1
typedef int v4i __attribute__((ext_vector_type(4)));
typedef __attribute__((address_space(1))) v4i* gptr_v4i;  // global int4*
typedef __attribute__((address_space(3))) v4i* lptr_v4i;  // LDS int4*
#else
#define HAVE_ASYNC_LDS 0
#endif

#if defined(__AMDGCN__)
#if __has_builtin(__builtin_amdgcn_s_wait_asynccnt)
#define WAIT_ASYNC(n) __builtin_amdgcn_s_wait_asynccnt(n)
#elif HAVE_ASYNC_LDS
#define WAIT_ASYNC(n) asm volatile("s_wait_asynccnt %0" ::"i"(n) : "memory")
#else
#define WAIT_ASYNC(n) ((void)0)
#endif
#define WAIT_DS0() asm volatile("s_wait_dscnt 0x0" ::: "memory")
#else
#define WAIT_ASYNC(n) ((void)0)
#define WAIT_DS0() ((void)0)
#endif

constexpr int kB = 64, kF = 256, kL = 2048;
constexpr int kRows = kB * kF;        // 16384 independent recurrences
constexpr int kT = 32;                // timesteps per chunk (full 128B lines/store)
constexpr int kQ = kT / 4;            // 8 float4s per chunk per lane
constexpr int kChunks = kL / kT;      // 64
constexpr int kRS = kT + 4;           // LDS row stride in floats (16B-aligned pad)
constexpr int kWaves = 2;             // 2 wave32 per block (keeps LDS < 64KB)
constexpr int kWG = kWaves * 32;      // 64 threads/block
constexpr int kStoresPerChunk = 3 * kQ;  // 24 async store instructions per chunk

// one LIF step: v_pre = a*v + (1-a)*x ; z = 15*(v_pre - .25) ; s = [v_pre>=.25] ; reset
#define LIF_STEP(xc, vc, zc, sc)                \
  {                                             \
    float vp = fmaf(alpha, v, oma * (xc));      \
    (vc) = vp;                                  \
    (zc) = fmaf(15.0f, vp, -3.75f);             \
    bool fire = vp >= 0.25f;                    \
    (sc) = fire ? 1.0f : 0.0f;                  \
    v = fire ? 0.0f : vp;                       \
  }

__global__ __launch_bounds__(kWG) void lif_fwd_kernel(
    const float* __restrict__ I, const float* __restrict__ raw_tau,
    float* __restrict__ out) {
  const int tid = threadIdx.x;
  const int lane = tid & 31;
  const int wave = tid >> 5;
  const int row = blockIdx.x * kWG + tid;  // one row (b,f) per thread

  // alpha = exp(-1 / (softplus(raw_tau[f]) + 1e-4)), f = row % 256
  float rt = raw_tau[row & (kF - 1)];
  float sp = (rt > 20.0f) ? rt : log1pf(expf(rt));
  float alpha = expf(-1.0f / (sp + 1e-4f));
  float oma = 1.0f - alpha;

  const float4* __restrict__ Irow =
      reinterpret_cast<const float4*>(I + (size_t)row * kL);
  const size_t N = (size_t)kRows * kL;
  float* const plane[3] = {out, out + N, out + 2 * N};  // v, z, s

  // per-wave private staging tiles: [wave][buf][plane][32 rows * kRS floats]
  __shared__ alignas(16) float lds[kWaves][2][3][32 * kRS];  // 55.3 KB total

  // register double-buffer for the input stream (8 b128 loads in flight)
  float4 cur[kQ], nxt[kQ];
#pragma unroll
  for (int q = 0; q < kQ; ++q) cur[q] = Irow[q];

  float v = 0.0f;
  for (int c = 0; c < kChunks; ++c) {
    // issue next chunk's loads before consuming this chunk (LOADcnt pipeline)
    if (c + 1 < kChunks) {
#pragma unroll
      for (int q = 0; q < kQ; ++q) nxt[q] = Irow[(c + 1) * kQ + q];
    }
    const int buf = c & 1;

    // kStoresPerChunk async stores issued per chunk; stores retire in order,
    // so <=kStoresPerChunk outstanding guarantees chunk c-2 (same buffer)
    // has fully drained before we overwrite it.
    WAIT_ASYNC(kStoresPerChunk);

    float* lv = &lds[wave][buf][0][lane * kRS];
    float* lz = &lds[wave][buf][1][lane * kRS];
    float* ls = &lds[wave][buf][2][lane * kRS];

#pragma unroll
    for (int q = 0; q < kQ; ++q) {
      float4 x = cur[q];
      float4 vo, zo, so;
      LIF_STEP(x.x, vo.x, zo.x, so.x)
      LIF_STEP(x.y, vo.y, zo.y, so.y)
      LIF_STEP(x.z, vo.z, zo.z, so.z)
      LIF_STEP(x.w, vo.w, zo.w, so.w)
      *reinterpret_cast<float4*>(lv + 4 * q) = vo;  // ds_store_b128
      *reinterpret_cast<float4*>(lz + 4 * q) = zo;
      *reinterpret_cast<float4*>(ls + 4 * q) = so;
    }

    // LDS data must be resident before the async engine reads it
    WAIT_DS0();

    // drain tile to global, time-coalesced: lanes cover (row_local, t4) with
    // t fastest -> 8 lanes x 16B contiguous per row = full 128B lines,
    // 4 rows per instruction.
#if HAVE_ASYNC_LDS
#pragma unroll
    for (int p = 0; p < 3; ++p) {
#pragma unroll
      for (int q = 0; q < kQ; ++q) {
        const int flat = q * 32 + lane;
        const int rloc = flat >> 3;  // 8 float4 per row
        const int t4 = flat & 7;
        const size_t grow = (size_t)blockIdx.x * kWG + wave * 32 + rloc;
        float* g = plane[p] + grow * kL + c * kT + t4 * 4;
        float* l = &lds[wave][buf][p][rloc * kRS + t4 * 4];
        __builtin_amdgcn_global_store_async_from_lds_b128(
            (gptr_v4i)g, (lptr_v4i)l, /*offset=*/0, /*cpol=*/0);
      }
    }
#else
#pragma unroll
    for (int p = 0; p < 3; ++p) {
#pragma unroll
      for (int q = 0; q < kQ; ++q) {
        const int flat = q * 32 + lane;
        const int rloc = flat >> 3;
        const int t4 = flat & 7;
        const size_t grow = (size_t)blockIdx.x * kWG + wave * 32 + rloc;
        float* g = plane[p] + grow * kL + c * kT + t4 * 4;
        const float* l = &lds[wave][buf][p][rloc * kRS + t4 * 4];
        *reinterpret_cast<float4*>(g) =
            *reinterpret_cast<const float4*>(l);
      }
    }
#endif

#pragma unroll
    for (int q = 0; q < kQ; ++q) cur[q] = nxt[q];
  }
  WAIT_ASYNC(0);
}

extern "C" void kernel_launch(void* const* d_in, const int* in_sizes, int n_in,
                              void* d_out, int out_size, void* d_ws,
                              size_t ws_size, hipStream_t stream) {
  const float* I = (const float*)d_in[0];        // [64,256,2048] f32
  const float* raw_tau = (const float*)d_in[1];  // [256] f32
  float* out = (float*)d_out;                    // v|z|s planes, each B*F*L

  dim3 grid(kRows / kWG);  // 256 blocks
  dim3 block(kWG);         // 64 threads = 2 wave32
  hipLaunchKernelGGL(lif_fwd_kernel, grid, block, 0, stream, I, raw_tau, out);
}